// DirectionalEncoding_48945447305222
// MI455X (gfx1250) — compile-verified
//
#include <hip/hip_runtime.h>

// Fused DirectionalEncoding for MI455X (gfx1250, wave32, WMMA).
// Single pass over HBM: dp in (25MB), out (128MB); all intermediates in
// registers/LDS. Both GEMMs run on V_WMMA_F32_16X16X4_F32 (exact fp32).
//
// Round-2 change: each lane computes only the 16 directions its lane-half
// supplies to the fp32-WMMA B-fragment (m = 4*(i/2)+(i&1)+2*hi) over the
// full K=16 neighborhood. This removes the shfl merge and, critically, all
// per-lane ternary selects from register arrays that previously lowered to
// v_cmp_eq/v_cndmask scan chains (~1.5K extra VALU).

typedef __attribute__((ext_vector_type(2))) float v2f;
typedef __attribute__((ext_vector_type(4))) float v4f;
typedef __attribute__((ext_vector_type(8))) float v8f;

#define B_   8
#define N_   16384
#define K_   16
#define M_   32      // directions
#define H_   128
#define O_   256
#define WAVES 8
#define NTILE_N (N_/16)        // 1024 column tiles per batch
#define NTILES  (B_*NTILE_N)   // 8192 total column tiles

// LDS layout (float offsets)
#define L_W1   0                       // 128*32  = 4096
#define L_W2   (L_W1 + H_*M_)          // 256*128 = 32768
#define L_BN   (L_W2 + O_*H_)          // scale/shift interleaved, 256
#define L_B2   (L_BN + 2*H_)           // 256
#define L_DIR  (L_B2 + O_)             // 3*32 padded to 128
#define L_H    (L_DIR + 128)           // per-wave 128x16 f32 staging, 8*2048
#define L_TOTAL (L_H + WAVES*H_*16)    // 53888 floats = 215552 B (<320KB WGP LDS)

extern "C" __global__ __launch_bounds__(256, 1)
void de_fused_kernel(const float* __restrict__ dp,
                     const float* __restrict__ dirv,
                     const float* __restrict__ w1,
                     const float* __restrict__ bn_gamma,
                     const float* __restrict__ bn_beta,
                     const float* __restrict__ bn_mean,
                     const float* __restrict__ bn_var,
                     const float* __restrict__ w2,
                     const float* __restrict__ b2,
                     float* __restrict__ out)
{
    extern __shared__ float lds[];
    const int tid  = threadIdx.x;
    const int lane = tid & 31;
    const int wv   = tid >> 5;
    const int col  = lane & 15;   // output column within 16-wide tile
    const int hi   = lane >> 4;   // lane half (fp32-WMMA K split)

    // ---- Stage weights / params into LDS (whole block cooperates) ----
    for (int i = tid; i < H_*M_; i += 256) lds[L_W1 + i] = w1[i];
    {
        const v4f* src = (const v4f*)w2;
        v4f*       dst = (v4f*)&lds[L_W2];
        #pragma unroll 4
        for (int i = tid; i < (O_*H_)/4; i += 256) dst[i] = src[i];
    }
    if (tid < H_) {
        float sc = bn_gamma[tid] * rsqrtf(bn_var[tid] + 1e-5f);
        lds[L_BN + 2*tid    ] = sc;
        lds[L_BN + 2*tid + 1] = bn_beta[tid] - bn_mean[tid] * sc;
    }
    if (tid < O_) lds[L_B2 + tid] = b2[tid];
    if (tid < M_) {
        float x = dirv[3*tid+0], y = dirv[3*tid+1], z = dirv[3*tid+2];
        // x/max(sqrt(s),1e-12) == x*rsqrt(max(s,1e-24))
        float inv = rsqrtf(fmaxf(x*x + y*y + z*z, 1e-24f));
        lds[L_DIR +      tid] = x*inv;
        lds[L_DIR + 32 + tid] = y*inv;
        lds[L_DIR + 64 + tid] = z*inv;
    }
    __syncthreads();

    // ---- Each wave processes one 16-column tile ----
    const int t  = blockIdx.x * WAVES + wv;   // 0..8191
    const int b  = t >> 10;
    const int n0 = (t & (NTILE_N - 1)) << 4;
    const int n  = n0 + col;

    // Per-lane direction subset: slot i <-> m = 4*(i/2) + (i&1) + 2*hi.
    // This is exactly the set of B-matrix rows this lane-half feeds to
    // V_WMMA_F32_16X16X4_F32 (B(k,n): VGPR k&1, lane n + 16*(k mod4 >= 2)).
    float dlx[16], dly[16], dlz[16];
    #pragma unroll
    for (int i = 0; i < 16; ++i) {
        int mi = 4*(i >> 1) + (i & 1) + 2*hi;
        dlx[i] = lds[L_DIR +      mi];
        dly[i] = lds[L_DIR + 32 + mi];
        dlz[i] = lds[L_DIR + 64 + mi];
    }

    float tm[16];
    #pragma unroll
    for (int i = 0; i < 16; ++i) tm[i] = -3.4e38f;

    const float* dpb = dp + (size_t)b*3*N_*K_ + (size_t)n*K_;
    #pragma unroll
    for (int k = 0; k < K_; ++k) {
        float x = dpb[k];
        float y = dpb[(size_t)N_*K_   + k];
        float z = dpb[(size_t)2*N_*K_ + k];
        float inv = rsqrtf(fmaxf(x*x + y*y + z*z, 1e-24f));
        x *= inv; y *= inv; z *= inv;
        #pragma unroll
        for (int i = 0; i < 16; ++i) {
            float th = fmaf(x, dlx[i], fmaf(y, dly[i], z*dlz[i]));
            tm[i] = fmaxf(tm[i], th);
        }
    }

    // ---- GEMM1: h(128x16) = W1(128x32) * theta_max(32x16), fp32 WMMA ----
    // fp32 WMMA frag layouts (16x16x4): A(m,k)->VGPR k&1, lane m+16*(k>=2);
    // B(k,n)->VGPR k&1, lane n+16*(k>=2); C/D row m -> VGPR m&7, lane-half m>=8.
    v8f hacc[8];
    #pragma unroll
    for (int r = 0; r < 8; ++r) hacc[r] = (v8f){};

    #pragma unroll
    for (int s = 0; s < 8; ++s) {                 // K-step: B rows 4s..4s+3
        v2f bfr;
        bfr.x = tm[2*s    ];                      // row 4s   + 2*hi
        bfr.y = tm[2*s + 1];                      // row 4s+1 + 2*hi
        #pragma unroll
        for (int r = 0; r < 8; ++r) {             // 8 independent acc chains
            v2f a = *(const v2f*)&lds[L_W1 + (r*16 + col)*M_ + 4*s + 2*hi];
            hacc[r] = __builtin_amdgcn_wmma_f32_16x16x4_f32(
                false, a, false, bfr, (short)0, hacc[r], false, false);
        }
    }

    // ---- BN + exact GELU on accumulator frags; stage h tile to LDS ----
    float* hbuf = &lds[L_H + wv * (H_*16)];
    #pragma unroll
    for (int r = 0; r < 8; ++r) {
        #pragma unroll
        for (int j = 0; j < 8; ++j) {
            int row = r*16 + j + hi*8;
            v2f p = *(const v2f*)&lds[L_BN + 2*row];   // (scale, shift)
            float yv = fmaf(hacc[r][j], p.x, p.y);
            float g  = 0.5f * yv * (1.0f + erff(yv * 0.70710678118654752f));
            hbuf[row*16 + col] = g;                    // per-wave region; DS in-order
        }
    }

    // Re-load h as B-fragments for GEMM2 (all 64 regs held live)
    float bh0[32], bh1[32];
    #pragma unroll
    for (int s = 0; s < 32; ++s) {
        int k0 = 4*s + 2*hi;
        bh0[s] = hbuf[ k0   *16 + col];
        bh1[s] = hbuf[(k0+1)*16 + col];
    }

    // ---- GEMM2: f(256x16) = W2(256x128) * h(128x16) + b2; store ----
    const size_t outbase = (size_t)b * O_ * N_ + n;
    for (int ot = 0; ot < 16; ot += 2) {          // 2 row-tiles -> 2 indep chains
        v8f fa = (v8f){}, fb = (v8f){};
        #pragma unroll
        for (int s = 0; s < 32; ++s) {
            v2f bfr; bfr.x = bh0[s]; bfr.y = bh1[s];
            v2f a0 = *(const v2f*)&lds[L_W2 + ((ot  )*16 + col)*H_ + 4*s + 2*hi];
            v2f a1 = *(const v2f*)&lds[L_W2 + ((ot+1)*16 + col)*H_ + 4*s + 2*hi];
            fa = __builtin_amdgcn_wmma_f32_16x16x4_f32(
                false, a0, false, bfr, (short)0, fa, false, false);
            fb = __builtin_amdgcn_wmma_f32_16x16x4_f32(
                false, a1, false, bfr, (short)0, fb, false, false);
        }
        #pragma unroll
        for (int j = 0; j < 8; ++j) {
            int row0 = ot*16 + j + hi*8;
            int row1 = row0 + 16;
            out[outbase + (size_t)row0*N_] = fa[j] + lds[L_B2 + row0];
            out[outbase + (size_t)row1*N_] = fb[j] + lds[L_B2 + row1];
        }
    }
}

extern "C" void kernel_launch(void* const* d_in, const int* in_sizes, int n_in,
                              void* d_out, int out_size, void* d_ws, size_t ws_size,
                              hipStream_t stream) {
    // setup_inputs order: dp, f0(unused), idx(unused), dir_vectors, w1,
    // bn_gamma, bn_beta, bn_mean, bn_var, w2, b2
    const float* dp    = (const float*)d_in[0];
    const float* dirv  = (const float*)d_in[3];
    const float* w1    = (const float*)d_in[4];
    const float* gam   = (const float*)d_in[5];
    const float* bet   = (const float*)d_in[6];
    const float* mean  = (const float*)d_in[7];
    const float* var   = (const float*)d_in[8];
    const float* w2    = (const float*)d_in[9];
    const float* b2    = (const float*)d_in[10];
    float* out = (float*)d_out;
    (void)in_sizes; (void)n_in; (void)out_size; (void)d_ws; (void)ws_size;

    size_t shmem = (size_t)L_TOTAL * sizeof(float);  // ~215.5 KB of 320 KB WGP LDS
    de_fused_kernel<<<NTILES / WAVES, 32 * WAVES, shmem, stream>>>(
        dp, dirv, w1, gam, bet, mean, var, w2, b2, out);
}